// PoseGNN_PyG_2181843386627
// MI455X (gfx1250) — compile-verified
//
#include <hip/hip_runtime.h>
#include <hip/hip_bf16.h>

typedef __attribute__((ext_vector_type(2))) float v2f;
typedef __attribute__((ext_vector_type(4))) float v4f;
typedef __attribute__((ext_vector_type(8))) float v8f;

#define NNODE 25
#define NEDGE 64
#define HID   128
#define OUTD  128
#define BPB   16          // batches per block -> exactly one WMMA M-tile
#define NPAIR (BPB / 2)   // slot pairs for packed-fp32 processing
#define ZROW  16          // floats per (pair,node) row: 6 x float2 + pad, 64B

static __device__ __forceinline__ v2f lo2(v4f q) { return __builtin_shufflevector(q, q, 0, 1); }
static __device__ __forceinline__ v2f hi2(v4f q) { return __builtin_shufflevector(q, q, 2, 3); }

__launch_bounds__(128)
__global__ void sage_fused_kernel(const float* __restrict__ x,
                                  const int*   __restrict__ edge_index,
                                  const float* __restrict__ W1l,
                                  const float* __restrict__ W1r,
                                  const float* __restrict__ b1,
                                  const float* __restrict__ W2l,
                                  const float* __restrict__ W2r,
                                  const float* __restrict__ b2,
                                  float* __restrict__ out)
{
    __shared__ int   ssrc[NEDGE];
    __shared__ int   sdst[NEDGE];
    __shared__ float sdeginv[NNODE];
    __shared__ float snodew[NNODE];           // sum_{e:src=n} deg_inv[dst_e]/25
    __shared__ float snw2[NNODE * 2];         // snodew duplicated as float2 pairs
    __shared__ float sA1[NNODE * 26];         // A1[n][m] = deg_inv[n]*count(m->n)
    // Z rows: [ag0 pair, ag1 pair, ag2 pair, x0 pair, x1 pair, x2 pair, pad x4]
    __align__(16) __shared__ float zbuf[NPAIR * NNODE * ZROW];   // 12.8 KB
    __shared__ float As[BPB][2 * HID];        // 16 x 256 WMMA A-matrix

    const int tid  = threadIdx.x;
    const int lane = tid & 31;
    const int wave = tid >> 5;
    const int bbase = blockIdx.x * BPB;

    // ---- per-thread layer-1 weights for column k = tid (registers, packed) ----
    const int k = tid;
    const v2f wl0 = { W1l[0 * HID + k], W1l[0 * HID + k] };
    const v2f wl1 = { W1l[1 * HID + k], W1l[1 * HID + k] };
    const v2f wl2 = { W1l[2 * HID + k], W1l[2 * HID + k] };
    const v2f wr0 = { W1r[0 * HID + k], W1r[0 * HID + k] };
    const v2f wr1 = { W1r[1 * HID + k], W1r[1 * HID + k] };
    const v2f wr2 = { W1r[2 * HID + k], W1r[2 * HID + k] };
    const float b1k = b1[k];
    const v2f b1k2 = { b1k, b1k };

    // ---- one-time block staging ----
    if (tid < NEDGE) {
        ssrc[tid] = edge_index[tid];
        sdst[tid] = edge_index[NEDGE + tid];
    }
    for (int i = tid; i < NNODE * 26; i += 128) sA1[i] = 0.0f;
    if (tid < NNODE) snodew[tid] = 0.0f;
    __syncthreads();

    if (tid < NNODE) {
        int c = 0;
        #pragma unroll
        for (int e = 0; e < NEDGE; ++e) c += (sdst[e] == tid);
        sdeginv[tid] = 1.0f / fmaxf((float)c, 1.0f);
    }
    __syncthreads();

    if (tid < NEDGE) {
        const float di = sdeginv[sdst[tid]];
        atomicAdd(&sA1[sdst[tid] * 26 + ssrc[tid]], di);
        atomicAdd(&snodew[ssrc[tid]], di * (1.0f / 25.0f));
    }
    // coalesced load of all 16 batches' x into the packed pair layout
    {
        const size_t base = (size_t)bbase * (NNODE * 3);
        for (int i = tid; i < BPB * NNODE * 3; i += 128) {
            const int slot = i / (NNODE * 3);
            const int r = i % (NNODE * 3);
            const int n = r / 3, d = r % 3;
            const int pair = slot >> 1, sub = slot & 1;
            zbuf[(pair * NNODE + n) * ZROW + 6 + 2 * d + sub] = x[base + i];
        }
    }
    __syncthreads();

    // ---- dense neighborhood aggregation (both slots of a pair at once) ----
    for (int i = tid; i < NPAIR * NNODE * 3; i += 128) {
        const int pair = i / (NNODE * 3);
        const int r = i % (NNODE * 3);
        const int n = r / 3, d = r % 3;
        v2f acc = { 0.0f, 0.0f };
        #pragma unroll
        for (int m = 0; m < NNODE; ++m) {
            const float a = sA1[n * 26 + m];
            const v2f xm = *(const v2f*)&zbuf[(pair * NNODE + m) * ZROW + 6 + 2 * d];
            acc = __builtin_elementwise_fma(xm, (v2f){ a, a }, acc);
        }
        *(v2f*)&zbuf[(pair * NNODE + n) * ZROW + 2 * d] = acc;
    }
    if (tid < NNODE) {
        const float w = snodew[tid];
        snw2[2 * tid] = w;
        snw2[2 * tid + 1] = w;
    }
    __syncthreads();

    // ---- fused layer-1 + column reductions, packed fp32 (2 slots / iter) ----
    for (int pair = 0; pair < NPAIR; ++pair) {
        const float* zp = &zbuf[pair * NNODE * ZROW];
        v2f cl = { 0.0f, 0.0f };
        v2f cr = { 0.0f, 0.0f };
        #pragma unroll
        for (int n = 0; n < NNODE; ++n) {
            const float* row = zp + n * ZROW;
            const v4f q0 = *(const v4f*)(row);      // ag d0 pair | ag d1 pair
            const v4f q1 = *(const v4f*)(row + 4);  // ag d2 pair | x  d0 pair
            const v4f q2 = *(const v4f*)(row + 8);  // x  d1 pair | x  d2 pair
            v2f pre = b1k2;
            pre = __builtin_elementwise_fma(lo2(q0), wl0, pre);
            pre = __builtin_elementwise_fma(hi2(q0), wl1, pre);
            pre = __builtin_elementwise_fma(lo2(q1), wl2, pre);
            pre = __builtin_elementwise_fma(hi2(q1), wr0, pre);
            pre = __builtin_elementwise_fma(lo2(q2), wr1, pre);
            pre = __builtin_elementwise_fma(hi2(q2), wr2, pre);
            const v2f h = __builtin_elementwise_max(pre, (v2f){ 0.0f, 0.0f });
            const v2f nw = *(const v2f*)&snw2[2 * n];
            cl = __builtin_elementwise_fma(h, nw, cl);
            cr = cr + h;
        }
        As[2 * pair + 0][k] = cl.x;
        As[2 * pair + 1][k] = cl.y;
        As[2 * pair + 0][HID + k] = cr.x * (1.0f / 25.0f);
        As[2 * pair + 1][HID + k] = cr.y * (1.0f / 25.0f);
    }
    __syncthreads();

    // ---- WMMA GEMM: (16 x 256) @ [W2l; W2r](256 x 128) + b2 -> out rows ----
    {
        const int hi  = lane >> 4;    // 0: K pair {0,1}, 1: K pair {2,3}
        const int l15 = lane & 15;
        const int n0a = wave * 32;
        const int n0b = n0a + 16;

        v8f c0, c1;
        const float bza = b2[n0a + l15];
        const float bzb = b2[n0b + l15];
        #pragma unroll
        for (int v = 0; v < 8; ++v) { c0[v] = bza; c1[v] = bzb; }

        const float* Wmats[2] = { W2l, W2r };
        #pragma unroll
        for (int half = 0; half < 2; ++half) {
            const float* W = Wmats[half];
            const int kofs = half * HID;
            #pragma unroll 4
            for (int k0 = 0; k0 < HID; k0 += 4) {
                const int kb = k0 + (hi << 1);
                v2f a;
                a.x = As[l15][kofs + kb];
                a.y = As[l15][kofs + kb + 1];
                v2f bfa, bfb;
                bfa.x = W[kb * OUTD + n0a + l15];
                bfa.y = W[(kb + 1) * OUTD + n0a + l15];
                bfb.x = W[kb * OUTD + n0b + l15];
                bfb.y = W[(kb + 1) * OUTD + n0b + l15];
                c0 = __builtin_amdgcn_wmma_f32_16x16x4_f32(
                        false, a, false, bfa, (short)0, c0, false, false);
                c1 = __builtin_amdgcn_wmma_f32_16x16x4_f32(
                        false, a, false, bfb, (short)0, c1, false, false);
            }
        }

        // D layout: lanes 0-15 -> M = v, N = lane; lanes 16-31 -> M = 8+v, N = lane-16
        #pragma unroll
        for (int v = 0; v < 8; ++v) {
            const int m = v + (hi << 3);
            out[(size_t)(bbase + m) * OUTD + n0a + l15] = c0[v];
            out[(size_t)(bbase + m) * OUTD + n0b + l15] = c1[v];
        }
    }
}

extern "C" void kernel_launch(void* const* d_in, const int* in_sizes, int n_in,
                              void* d_out, int out_size, void* d_ws, size_t ws_size,
                              hipStream_t stream) {
    const float* x   = (const float*)d_in[0];
    const int*   ei  = (const int*)d_in[1];
    const float* W1l = (const float*)d_in[2];
    const float* W1r = (const float*)d_in[3];
    const float* b1  = (const float*)d_in[4];
    const float* W2l = (const float*)d_in[5];
    const float* W2r = (const float*)d_in[6];
    const float* b2  = (const float*)d_in[7];
    float* out = (float*)d_out;

    const int B = in_sizes[0] / (NNODE * 3);   // 16384
    const int blocks = B / BPB;                // 1024
    sage_fused_kernel<<<blocks, 128, 0, stream>>>(x, ei, W1l, W1r, b1, W2l, W2r, b2, out);
}